// InnerMemoryLayer_9070970929596
// MI455X (gfx1250) — compile-verified
//
#include <hip/hip_runtime.h>

// ---- problem constants (from reference) ----
constexpr int kT = 256;    // timesteps
constexpr int kU = 2048;   // units (rows)
constexpr int kD = 2048;   // last_dim (cols)

// ---- tiling ----
constexpr int ROWS    = 2;   // u-rows per block (keeps VGPRs < 256, no MSB toggles)
constexpr int EPT     = 8;   // d-elements per thread per row
constexpr int THREADS = 256; // 8 waves (wave32)

typedef float v2f __attribute__((ext_vector_type(2)));
typedef float v8f __attribute__((ext_vector_type(8)));

// Raw v_exp_f32 (no libm denormal guard). Safe here: the argument is
// g2*x in (-2, 0], far from the subnormal threshold near -126.
__device__ __forceinline__ float fast_exp2(float x) {
    return __builtin_amdgcn_exp2f(x);
}

// Full 32-lane sum on the matrix pipe (layout-immune all-ones B):
//   A(16x4): A[m,0]=p[m], A[m,2]=p[m+16], a.y=0 ; B(4x16)=ones
//   -> D[m,n] = p[m]+p[m+16]; per-lane sum of the 8 D VGPRs gives the two
//   half-sums, xor-16 shuffle completes the reduction (result in all lanes).
__device__ __forceinline__ float wave_reduce_wmma(float p) {
    v2f a; a.x = p;    a.y = 0.0f;
    v2f b; b.x = 1.0f; b.y = 1.0f;
    v8f c = {};
    c = __builtin_amdgcn_wmma_f32_16x16x4_f32(
            /*neg_a=*/false, a, /*neg_b=*/false, b,
            /*c_mod=*/(short)0, c, /*reuse_a=*/false, /*reuse_b=*/false);
    float s = ((c[0] + c[1]) + (c[2] + c[3])) + ((c[4] + c[5]) + (c[6] + c[7]));
    s += __shfl_xor(s, 16, 32);
    return s;
}

__global__ void __launch_bounds__(THREADS)
inner_memory_layer_scan(const float* __restrict__ inp,     // (T, D)
                        const float* __restrict__ weight,  // (U, D)
                        const float* __restrict__ augment, // (U, D)
                        const float* __restrict__ bias,    // (U,)
                        const float* __restrict__ fup,     // (U, D)
                        const float* __restrict__ flow,    // (U, D)
                        const float* __restrict__ res0,    // (U, D)
                        const float* __restrict__ fac0,    // (U, D)
                        const float* __restrict__ rah,     // (U, D)
                        float* __restrict__ out)           // (T, U)
{
    const int tid  = threadIdx.x;
    const int lane = tid & 31;
    const int wave = tid >> 5;
    const int u0   = blockIdx.x * ROWS;
    const int d0   = tid * EPT;

    // Register-resident per-element constants + state (6 floats x 16 elements).
    // g2 = -log2(e)*exp(augment): folds base-e->base-2 conversion AND the
    // negation into the step-invariant constant, so each step is mul+exp2 only.
    float g2[ROWS][EPT], ew[ROWS][EPT], fl[ROWS][EPT], dfu[ROWS][EPT];
    float res[ROWS][EPT], fac[ROWS][EPT];

    constexpr float kNegLog2e = -1.44269504088896340736f;

    #pragma unroll
    for (int r = 0; r < ROWS; ++r) {
        const int base = (u0 + r) * kD + d0;
        #pragma unroll
        for (int j = 0; j < EPT; ++j) {
            const float w  = weight[base + j];
            const float au = augment[base + j];
            const float up = fup[base + j];
            const float lo = flow[base + j];
            const float rh = rah[base + j];
            g2[r][j]  = kNegLog2e * __expf(au);            // -log2e * gain
            ew[r][j]  = w + fabsf(w) * rh;                 // eff_w
            const float sfu = 0.5f / (1.0f + __expf(-up)); // 0.5*sigmoid(fu)
            const float sfl = sfu / (1.0f + __expf(-lo));  // fu*sigmoid(fl)
            fl[r][j]  = sfl;
            dfu[r][j] = sfu - sfl;                         // fu - fl
            res[r][j] = res0[base + j];
            fac[r][j] = fac0[base + j];
        }
    }

    // bias for the final store lanes (wave 0, 8 lanes per row)
    float biasv = 0.0f;
    if (tid < ROWS * 8) biasv = bias[u0 + (tid >> 3)];

    __shared__ float red[2][ROWS][8]; // double-buffered per-wave partials

    for (int t = 0; t < kT; ++t) {
        // This step's input slice (8 contiguous floats per thread, L2-resident)
        float inpv[EPT];
        const float* ip = inp + t * kD + d0;
        *(float4*)&inpv[0] = *(const float4*)&ip[0];
        *(float4*)&inpv[4] = *(const float4*)&ip[4];
        if (t + 1 < kT) __builtin_prefetch(inp + (t + 1) * kD + d0, 0, 0);

        float ps[ROWS];
        #pragma unroll
        for (int r = 0; r < ROWS; ++r) {
            float p = 0.0f;
            #pragma unroll
            for (int j = 0; j < EPT; ++j) {
                const float e = fast_exp2(g2[r][j] * inpv[j]); // exp(-gain*inp) = 1-activity
                const float a = 1.0f - e;                      // activity
                p = fmaf(a * ew[r][j], res[r][j], p);          // output contribution
                const float rest = fmaf(fac[r][j], dfu[r][j], fl[r][j]);
                const float nr   = res[r][j] * e;              // resource*(1-activity)
                res[r][j] = fmaf(1.0f - nr, rest, nr);         // nr + (1-nr)*rest
                const float nf = fmaf(-0.1f * rest, fac[r][j], fac[r][j]); // fac*(1-0.1*rest)
                fac[r][j] = fmaf(nf, e, a);                    // nf + (1-nf)*a == a + nf*e
            }
            ps[r] = p;
        }

        // In-wave reduction on the matrix pipe (uniform code: EXEC all-ones).
        #pragma unroll
        for (int r = 0; r < ROWS; ++r) {
            const float s = wave_reduce_wmma(ps[r]);
            if (lane == 0) red[t & 1][r][wave] = s;
        }
        __syncthreads();

        // Wave 0 (lanes 0..15) combines the 8 per-wave partials per row.
        if (tid < ROWS * 8) {
            float v = red[t & 1][tid >> 3][tid & 7];
            v += __shfl_xor(v, 1, 32);
            v += __shfl_xor(v, 2, 32);
            v += __shfl_xor(v, 4, 32);
            if ((tid & 7) == 0) {
                out[t * kU + u0 + (tid >> 3)] = fmaxf(v + biasv, 0.0f); // relu(sum+bias)
            }
        }
        // Single barrier per step is safe: double buffering + the next step's
        // barrier orders wave0's reads of red[t&1] before its reuse at t+2.
    }
}

extern "C" void kernel_launch(void* const* d_in, const int* in_sizes, int n_in,
                              void* d_out, int out_size, void* d_ws, size_t ws_size,
                              hipStream_t stream) {
    const float* inp     = (const float*)d_in[0];
    const float* weight  = (const float*)d_in[1];
    const float* augment = (const float*)d_in[2];
    const float* bias    = (const float*)d_in[3];
    const float* fup     = (const float*)d_in[4];
    const float* flow    = (const float*)d_in[5];
    const float* res0    = (const float*)d_in[6];
    const float* fac0    = (const float*)d_in[7];
    // d_in[8] (ra_mean0) and d_in[9] (ra_var0) do not affect the returned outputs.
    const float* rah     = (const float*)d_in[10];
    float* out = (float*)d_out;

    dim3 grid(kU / ROWS);   // 1024 blocks
    dim3 block(THREADS);    // 256 threads = 8 wave32
    hipLaunchKernelGGL(inner_memory_layer_scan, grid, block, 0, stream,
                       inp, weight, augment, bias, fup, flow, res0, fac0, rah, out);
}